// GNNSegmentClassifierG_67937792688144
// MI455X (gfx1250) — compile-verified
//
#include <hip/hip_runtime.h>
#include <math.h>

typedef __attribute__((ext_vector_type(2))) float v2f;
typedef __attribute__((ext_vector_type(8))) float v8f;

// ---------------------------------------------------------------------------
// xcat[n] = [ tanh(X[n] @ win_w + win_b) (8), X[n] (2) ]   -> [N,10]
// ---------------------------------------------------------------------------
__global__ void k_init_xcat(const float* __restrict__ X,
                            const float* __restrict__ win_w,   // [2,8]
                            const float* __restrict__ win_b,   // [8]
                            float* __restrict__ xcat, int N) {
    int n = blockIdx.x * blockDim.x + threadIdx.x;
    if (n >= N) return;
    float x0 = X[2 * n], x1 = X[2 * n + 1];
    float* o = xcat + (size_t)n * 10;
#pragma unroll
    for (int j = 0; j < 8; ++j)
        o[j] = tanhf(x0 * win_w[j] + x1 * win_w[8 + j] + win_b[j]);
    o[8] = x0;
    o[9] = x1;
}

__global__ void k_zero(float* __restrict__ p, int n) {
    int i = blockIdx.x * blockDim.x + threadIdx.x;
    if (i < n) p[i] = 0.0f;
}

// ---------------------------------------------------------------------------
// Fused edge net (+ optional scatter):
//   B = [xcat[col], xcat[row]]  (16 edges x 20 feats per wave, staged in LDS)
//   h = tanh(B @ e1_w + e1_b)   via 5x V_WMMA_F32_16X16X4_F32 (full fp32)
//   e = sigmoid(h @ e2_w + e2_b) via per-lane mul + shfl_xor reduction
//   mode 0: mi[col] += xcat[row]*e ; mo[row] += xcat[col]*e  (atomics)
//   mode 1: out[edge] = e
// ---------------------------------------------------------------------------
__global__ __launch_bounds__(256) void k_edge(
    const float* __restrict__ xcat,
    const int*   __restrict__ eidx,   // [2,E] : row = eidx[0..E), col = eidx[E..2E)
    const float* __restrict__ e1_w,   // [20,8]
    const float* __restrict__ e1_b,   // [8]
    const float* __restrict__ e2_w,   // [8,1]
    const float* __restrict__ e2_b,   // [1]
    float* __restrict__ mi,           // [N,10] (mode 0)
    float* __restrict__ mo,           // [N,10] (mode 0)
    float* __restrict__ out,          // [E]    (mode 1)
    int E, int mode) {
    __shared__ float smem[8 * 336];               // per wave: 16x20 tile + 16 e-values
    const int lane = threadIdx.x & 31;
    const int wave = threadIdx.x >> 5;
    const int m  = lane & 15;                     // edge-in-tile / WMMA row
    const int hi = lane >> 4;                     // half-wave selector
    float* tile  = smem + wave * 336;             // [16][20]
    float* etile = tile + 320;                    // [16]

    long tileBase = (long)(blockIdx.x * 8 + wave) * 16;
    if (tileBase >= (long)E) return;              // wave-uniform exit (EXEC stays full)
    int  j      = (int)tileBase + m;
    bool active = j < E;
    if (!active) j = E - 1;                       // clamp (select, no divergence)

    int ri = eidx[j];                             // row index
    int ci = eidx[E + j];                         // col index
    int node = hi ? ri : ci;                      // lo lanes: x[col]; hi lanes: x[row]
    const float* src = xcat + (size_t)node * 10;
    float* dst = tile + m * 20 + hi * 10;
    v2f d[5];
#pragma unroll
    for (int k = 0; k < 5; ++k) {                 // gather 10 feats -> regs + LDS tile
        d[k] = *(const v2f*)(src + 2 * k);
        *(v2f*)(dst + 2 * k) = d[k];
    }

    // e1_w B-fragments: B is KxN = 4x16; lane n = m, K rows 2*hi, 2*hi+1 per chunk.
    const int  n  = m;
    const bool nv = n < 8;                        // N padded 8 -> 16 with zeros
    v2f wb[5];
#pragma unroll
    for (int c = 0; c < 5; ++c) {
        int k0 = 4 * c + 2 * hi;
        wb[c].x = nv ? e1_w[k0 * 8 + n] : 0.0f;
        wb[c].y = nv ? e1_w[(k0 + 1) * 8 + n] : 0.0f;
    }
    float bN  = nv ? e1_b[n] : 0.0f;
    float w2N = nv ? e2_w[n] : 0.0f;
    float b2  = e2_b[0];

    // A-fragment per chunk: lane holds A[m][2*hi], A[m][2*hi+1] (16x4 f32 layout).
    v8f acc = {};
#pragma unroll
    for (int c = 0; c < 5; ++c) {
        v2f a = *(const v2f*)(tile + m * 20 + 4 * c + 2 * hi);
        acc = __builtin_amdgcn_wmma_f32_16x16x4_f32(false, a, false, wb[c],
                                                    (short)0, acc, false, false);
    }

    // Second layer: acc[r] = pre-act of edge (r + 8*hi), feature n = lane%16.
    // Reduce over n within each 16-lane half.
    float ev[8];
#pragma unroll
    for (int r = 0; r < 8; ++r) {
        float t = tanhf(acc[r] + bN) * w2N;       // zero for padded n>=8
        t += __shfl_xor(t, 1, 32);
        t += __shfl_xor(t, 2, 32);
        t += __shfl_xor(t, 4, 32);
        t += __shfl_xor(t, 8, 32);
        ev[r] = 1.0f / (1.0f + expf(-(t + b2)));
    }
    // Broadcast e per edge via LDS (same-wave LDS is in-order; no barrier needed).
    if (m == 0) {                                 // lanes 0 and 16
#pragma unroll
        for (int r = 0; r < 8; ++r) etile[hi * 8 + r] = ev[r];
    }
    float e = etile[m];                           // e for this lane's edge

    if (mode == 0) {
        if (active) {
            // hi lanes hold xcat[row] -> mi[col];  lo lanes hold xcat[col] -> mo[row]
            float* base = hi ? (mi + (size_t)ci * 10) : (mo + (size_t)ri * 10);
#pragma unroll
            for (int k = 0; k < 5; ++k) {
                __hip_atomic_fetch_add(base + 2 * k,     d[k].x * e,
                                       __ATOMIC_RELAXED, __HIP_MEMORY_SCOPE_AGENT);
                __hip_atomic_fetch_add(base + 2 * k + 1, d[k].y * e,
                                       __ATOMIC_RELAXED, __HIP_MEMORY_SCOPE_AGENT);
            }
        }
    } else {
        if (active && hi == 0) out[j] = e;        // lanes 0..15 cover all 16 edges
    }
}

// ---------------------------------------------------------------------------
// Node net: M = [mi, mo, xcat] (30) -> tanh -> tanh -> xcat[:,0:8]
// ---------------------------------------------------------------------------
__global__ void k_node(float* __restrict__ xcat,
                       const float* __restrict__ mi,
                       const float* __restrict__ mo,
                       const float* __restrict__ n1_w,  // [30,8]
                       const float* __restrict__ n1_b,  // [8]
                       const float* __restrict__ n2_w,  // [8,8]
                       const float* __restrict__ n2_b,  // [8]
                       int N) {
    int nd = blockIdx.x * blockDim.x + threadIdx.x;
    if (nd >= N) return;
    float M[30];
#pragma unroll
    for (int k = 0; k < 10; ++k) M[k]      = mi[(size_t)nd * 10 + k];
#pragma unroll
    for (int k = 0; k < 10; ++k) M[10 + k] = mo[(size_t)nd * 10 + k];
#pragma unroll
    for (int k = 0; k < 10; ++k) M[20 + k] = xcat[(size_t)nd * 10 + k];
    float h[8];
#pragma unroll
    for (int jj = 0; jj < 8; ++jj) {
        float s = n1_b[jj];
#pragma unroll
        for (int k = 0; k < 30; ++k) s += M[k] * n1_w[k * 8 + jj];
        h[jj] = tanhf(s);
    }
#pragma unroll
    for (int jj = 0; jj < 8; ++jj) {
        float s = n2_b[jj];
#pragma unroll
        for (int k = 0; k < 8; ++k) s += h[k] * n2_w[k * 8 + jj];
        xcat[(size_t)nd * 10 + jj] = tanhf(s);
    }
}

extern "C" void kernel_launch(void* const* d_in, const int* in_sizes, int n_in,
                              void* d_out, int out_size, void* d_ws, size_t ws_size,
                              hipStream_t stream) {
    (void)n_in; (void)out_size; (void)ws_size;
    const float* X     = (const float*)d_in[0];
    const int*   eidx  = (const int*)d_in[1];
    const float* win_w = (const float*)d_in[2];
    const float* win_b = (const float*)d_in[3];
    const float* e1_w  = (const float*)d_in[4];
    const float* e1_b  = (const float*)d_in[5];
    const float* e2_w  = (const float*)d_in[6];
    const float* e2_b  = (const float*)d_in[7];
    const float* n1_w  = (const float*)d_in[8];
    const float* n1_b  = (const float*)d_in[9];
    const float* n2_w  = (const float*)d_in[10];
    const float* n2_b  = (const float*)d_in[11];
    float* out = (float*)d_out;

    const int N = in_sizes[0] / 2;   // IN_DIM = 2
    const int E = in_sizes[1] / 2;   // edge_index is [2,E]

    // Workspace: xcat [N,10] | mi [N,10] | mo [N,10]  (mi,mo contiguous) ~60MB
    float* xcat = (float*)d_ws;
    float* mi   = xcat + (size_t)N * 10;
    float* mo   = mi   + (size_t)N * 10;

    const int tb    = 256;
    const int gN    = (N + tb - 1) / tb;
    const int tiles = (E + 15) / 16;
    const int gE    = (tiles + 7) / 8;           // 8 waves (tiles) per block
    const int zn    = N * 20;                    // zero mi+mo together
    const int gZ    = (zn + tb - 1) / tb;

    k_init_xcat<<<gN, tb, 0, stream>>>(X, win_w, win_b, xcat, N);
    for (int it = 0; it < 3; ++it) {
        k_zero<<<gZ, tb, 0, stream>>>(mi, zn);
        k_edge<<<gE, tb, 0, stream>>>(xcat, eidx, e1_w, e1_b, e2_w, e2_b,
                                      mi, mo, out, E, 0);
        k_node<<<gN, tb, 0, stream>>>(xcat, mi, mo, n1_w, n1_b, n2_w, n2_b, N);
    }
    k_edge<<<gE, tb, 0, stream>>>(xcat, eidx, e1_w, e1_b, e2_w, e2_b,
                                  mi, mo, out, E, 1);
}